// FuseLayer_15556371546958
// MI455X (gfx1250) — compile-verified
//
#include <hip/hip_runtime.h>
#include <hip/hip_bf16.h>
#include <math.h>

// ---------------------------------------------------------------------------
// MI455X (gfx1250) fused implementation.
// Bandwidth-bound problem (~130MB traffic, ~11 GFLOP): stream u once, fuse
// per-16-token tiles, all GEMMs via V_WMMA_F32_16X16X4_F32 (fp32-exact).
// ---------------------------------------------------------------------------

typedef float v2f __attribute__((ext_vector_type(2)));
typedef float v8f __attribute__((ext_vector_type(8)));

__device__ __forceinline__ v8f wmma4(v2f a, v2f b, v8f c) {
    // D = A(16x4,f32) * B(4x16,f32) + C(16x16,f32)
    return __builtin_amdgcn_wmma_f32_16x16x4_f32(
        /*neg_a=*/false, a, /*neg_b=*/false, b,
        /*c_mod=*/(short)0, c, /*reuse_a=*/false, /*reuse_b=*/false);
}

// Generic small transpose: out[c*R + r] = in[r*C + c]   (in is [R][C])
__global__ void k_transpose(const float* __restrict__ in, float* __restrict__ out,
                            int R, int C) {
    int i = blockIdx.x * blockDim.x + threadIdx.x;
    if (i < R * C) {
        int r = i / C, c = i % C;
        out[c * R + r] = in[i];
    }
}

// ---------------------------------------------------------------------------
// Kernel 1: grouped attention + shortcut. One block = 16 x-tokens.
// ---------------------------------------------------------------------------
__global__ __launch_bounds__(256) void k_attn(
    const float* __restrict__ u, const float* __restrict__ x,
    const float* __restrict__ WkT, const float* __restrict__ WqT,
    const float* __restrict__ bq,  const float* __restrict__ WvT,
    const float* __restrict__ bv,  const float* __restrict__ WsT,
    const float* __restrict__ bs,
    float* __restrict__ zbuf, float* __restrict__ shortcut)
{
    __shared__ float q_lds[16 * 64];       // q incl. bias, [n][64]
    __shared__ float attn_lds[16 * 4 * 16]; // [n][g][t], stride 16
    __shared__ float out_lds[16 * 64];     // output accum, seeded with bv

    const int tid  = threadIdx.x;
    const int wave = tid >> 5;
    const int lane = tid & 31;
    const int lm   = lane & 15;   // M / N index inside tile
    const int lh   = lane >> 4;   // half select

    const int blk = blockIdx.x;
    const int b   = blk / 448;
    const int n0g = (blk % 448) * 16;      // global token start in [0,7168)

    // bunch selection (static sizes: 4096 /2048 /1024 tokens, to = 4/8/16)
    int to, lg, nl0, p0, p1;
    if (n0g < 4096)      { to = 4;  lg = 2; nl0 = n0g;        p0 = 0;     p1 = 24576; }
    else if (n0g < 6144) { to = 8;  lg = 3; nl0 = n0g - 4096; p0 = 8192;  p1 = 32768; }
    else                 { to = 16; lg = 4; nl0 = n0g - 6144; p0 = 16384; p1 = 40960; }

    // u rows for this tile are contiguous: flat = n_local*to + t, split at 8192
    const int  f0    = nl0 * to;
    const long urow0 = (long)((f0 < 8192) ? (p0 + f0) : (p1 + (f0 - 8192)));
    const float* ubase = u + ((long)b * 49152 + urow0) * 64;
    const float* xbase = x + ((long)b * 7168 + n0g) * 64;
    const long   zrow0 = (long)b * 7168 + n0g;

    // ---- Pass 0: q = x@Wq + bq (waves 0-3) ; shortcut = x@Ws + bs (waves 4-7)
    {
        const int  g        = wave & 3;
        const bool do_short = (wave >= 4);        // wave-uniform
        const float* WT   = do_short ? WsT : WqT;
        const float* bias = do_short ? bs  : bq;
        const int col = g * 16 + lm;
        v8f acc = {};
        const float* arow = xbase + lm * 64 + 2 * lh;
        const float* brow = WT + col * 64 + 2 * lh;
#pragma unroll
        for (int kb = 0; kb < 64; kb += 4)
            acc = wmma4(*(const v2f*)(arow + kb), *(const v2f*)(brow + kb), acc);
        float accs[8];
#pragma unroll
        for (int r = 0; r < 8; ++r) accs[r] = acc[r];
        const float bcol = bias[col];
#pragma unroll
        for (int r = 0; r < 8; ++r) {
            const int M = r + 8 * lh;
            const float val = accs[r] + bcol;
            if (do_short) shortcut[(zrow0 + M) * 64 + col] = val;
            else          q_lds[M * 64 + col] = val;
        }
    }
    __syncthreads();

    // ---- k pass: logits = (u@Wk) . q * 1/sqrt(64)   (bk is softmax-invariant)
    const int njobs = 4 * to;  // (16*to/16) row tiles x 4 groups
    for (int job = wave; job < njobs; job += 8) {
        const int ut0 = (job >> 2) * 16;
        const int g   = job & 3;
        v8f acc = {};
        const float* arow = ubase + (ut0 + lm) * 64 + 2 * lh;
        const float* brow = WkT + (g * 16 + lm) * 64 + 2 * lh;
#pragma unroll
        for (int kb = 0; kb < 64; kb += 4)
            acc = wmma4(*(const v2f*)(arow + kb), *(const v2f*)(brow + kb), acc);
        float accs[8];
#pragma unroll
        for (int r = 0; r < 8; ++r) accs[r] = acc[r];
#pragma unroll
        for (int r = 0; r < 8; ++r) {
            const int ut = ut0 + r + 8 * lh;          // u-token index (local)
            const int nl = ut >> lg;                  // x-token (local)
            float pv = accs[r] * q_lds[nl * 64 + g * 16 + lm];
            pv += __shfl_xor(pv, 1, 16);
            pv += __shfl_xor(pv, 2, 16);
            pv += __shfl_xor(pv, 4, 16);
            pv += __shfl_xor(pv, 8, 16);
            if (lm == 0)
                attn_lds[(nl * 4 + g) * 16 + (ut & (to - 1))] = pv * 0.125f;
        }
    }
    __syncthreads();

    // ---- softmax over t per (n,g); seed out with bv (softmax sums to 1)
    if (tid < 64) {
        float* row = &attn_lds[tid * 16];
        float mx = -INFINITY;
        for (int t = 0; t < to; ++t) mx = fmaxf(mx, row[t]);
        float s = 0.f;
        for (int t = 0; t < to; ++t) { float e = __expf(row[t] - mx); row[t] = e; s += e; }
        const float inv = 1.0f / s;
        for (int t = 0; t < to; ++t) row[t] *= inv;
    }
    for (int idx = tid; idx < 1024; idx += 256) out_lds[idx] = bv[idx & 63];
    __syncthreads();

    // ---- v pass: out += attn * (u@Wv). Collision-free LDS writes (no atomics)
    for (int job = wave; job < njobs; job += 8) {
        const int ut0 = (job >> 2) * 16;
        const int g   = job & 3;
        v8f acc = {};
        const float* arow = ubase + (ut0 + lm) * 64 + 2 * lh;
        const float* brow = WvT + (g * 16 + lm) * 64 + 2 * lh;
#pragma unroll
        for (int kb = 0; kb < 64; kb += 4)
            acc = wmma4(*(const v2f*)(arow + kb), *(const v2f*)(brow + kb), acc);
        float accs[8];
#pragma unroll
        for (int r = 0; r < 8; ++r) accs[r] = acc[r];
        const int col = g * 16 + lm;
        if (to == 16) {                               // whole tile is one n
            float sum = 0.f;
#pragma unroll
            for (int r = 0; r < 8; ++r) {
                const int ut = ut0 + r + 8 * lh;
                sum += attn_lds[((ut >> 4) * 4 + g) * 16 + (ut & 15)] * accs[r];
            }
            sum += __shfl_xor(sum, 16, 32);           // combine halves
            if (lh == 0) out_lds[(ut0 >> 4) * 64 + col] += sum;
        } else {                                      // halves hit distinct n
            float sum = 0.f;
            int cur = (ut0 + 8 * lh) >> lg;
#pragma unroll
            for (int r = 0; r < 8; ++r) {
                const int ut = ut0 + r + 8 * lh;
                const int nl = ut >> lg;
                if (nl != cur) { out_lds[cur * 64 + col] += sum; sum = 0.f; cur = nl; }
                sum += attn_lds[(nl * 4 + g) * 16 + (ut & (to - 1))] * accs[r];
            }
            out_lds[cur * 64 + col] += sum;
        }
    }
    __syncthreads();

    for (int idx = tid; idx < 1024; idx += 256)
        zbuf[zrow0 * 64 + idx] = out_lds[idx];
}

// ---------------------------------------------------------------------------
// Kernel 2: LN1 -> gelu(z@Wm1+bm1)@Wm2+bm2 -> LN2 + shortcut
// ---------------------------------------------------------------------------
__global__ __launch_bounds__(256) void k_mlp(
    const float* __restrict__ zbuf, const float* __restrict__ shortcut,
    const float* __restrict__ g1, const float* __restrict__ be1,
    const float* __restrict__ Wm1T, const float* __restrict__ bm1,
    const float* __restrict__ Wm2T, const float* __restrict__ bm2,
    const float* __restrict__ g2, const float* __restrict__ be2,
    float* __restrict__ out)
{
    __shared__ float ln_lds[16 * 64];
    __shared__ float h_lds[16 * 256];
    __shared__ float m_a[16 * 64];   // K-split partial 0
    __shared__ float m_b[16 * 64];   // K-split partial 1
    __shared__ float mu[16], rs[16];

    const int tid  = threadIdx.x;
    const int wave = tid >> 5;
    const int lane = tid & 31;
    const int lm   = lane & 15;
    const int lh   = lane >> 4;
    const long row0 = (long)blockIdx.x * 16;

    for (int idx = tid; idx < 1024; idx += 256) ln_lds[idx] = zbuf[row0 * 64 + idx];
    __syncthreads();

    // LN1 (in place)
    if (tid < 16) {
        float* row = &ln_lds[tid * 64];
        float m = 0.f;
        for (int c = 0; c < 64; ++c) m += row[c];
        m *= (1.f / 64.f);
        float v = 0.f;
        for (int c = 0; c < 64; ++c) { float d = row[c] - m; v += d * d; }
        v *= (1.f / 64.f);
        const float r = rsqrtf(v + 1e-5f);
        for (int c = 0; c < 64; ++c) row[c] = (row[c] - m) * r * g1[c] + be1[c];
    }
    __syncthreads();

    // GEMM1: h = gelu(ln @ Wm1 + bm1), N=256 -> 16 col tiles, 2 per wave
#pragma unroll
    for (int jj = 0; jj < 2; ++jj) {
        const int ct  = wave + jj * 8;
        const int col = ct * 16 + lm;
        v8f acc = {};
        const float* brow = Wm1T + col * 64 + 2 * lh;
#pragma unroll
        for (int kb = 0; kb < 64; kb += 4) {
            v2f a = *(const v2f*)(&ln_lds[lm * 64 + 2 * lh + kb]);
            acc = wmma4(a, *(const v2f*)(brow + kb), acc);
        }
        float accs[8];
#pragma unroll
        for (int r = 0; r < 8; ++r) accs[r] = acc[r];
        const float bcol = bm1[col];
#pragma unroll
        for (int r = 0; r < 8; ++r) {
            const int M = r + 8 * lh;
            const float xg = accs[r] + bcol;
            // exact GELU: 0.5*x*(1+erf(x/sqrt(2)))
            h_lds[M * 256 + col] = 0.5f * xg * (1.0f + erff(xg * 0.70710678118654752f));
        }
    }
    __syncthreads();

    // GEMM2: m = h @ Wm2, N=64 -> 4 col tiles, K=256 split in halves over 8 waves
    {
        const int ct  = wave & 3;
        const int ks  = (wave >> 2) * 128;
        const int col = ct * 16 + lm;
        float* mpart = (wave < 4) ? m_a : m_b;
        v8f acc = {};
        const float* arow = &h_lds[lm * 256 + ks + 2 * lh];
        const float* brow = Wm2T + col * 256 + ks + 2 * lh;
#pragma unroll
        for (int kb = 0; kb < 128; kb += 4)
            acc = wmma4(*(const v2f*)(arow + kb), *(const v2f*)(brow + kb), acc);
        float accs[8];
#pragma unroll
        for (int r = 0; r < 8; ++r) accs[r] = acc[r];
#pragma unroll
        for (int r = 0; r < 8; ++r)
            mpart[(r + 8 * lh) * 64 + col] = accs[r];
    }
    __syncthreads();

    // LN2 stats (combine partials + bm2 deterministically)
    if (tid < 16) {
        float* ra = &m_a[tid * 64];
        const float* rb = &m_b[tid * 64];
        float m = 0.f;
        for (int c = 0; c < 64; ++c) { float t = ra[c] + rb[c] + bm2[c]; ra[c] = t; m += t; }
        m *= (1.f / 64.f);
        float v = 0.f;
        for (int c = 0; c < 64; ++c) { float d = ra[c] - m; v += d * d; }
        v *= (1.f / 64.f);
        mu[tid] = m;
        rs[tid] = rsqrtf(v + 1e-5f);
    }
    __syncthreads();

    for (int idx = tid; idx < 1024; idx += 256) {
        const int n = idx >> 6, c = idx & 63;
        out[row0 * 64 + idx] =
            (m_a[idx] - mu[n]) * rs[n] * g2[c] + be2[c] + shortcut[row0 * 64 + idx];
    }
}

// ---------------------------------------------------------------------------
extern "C" void kernel_launch(void* const* d_in, const int* in_sizes, int n_in,
                              void* d_out, int out_size, void* d_ws, size_t ws_size,
                              hipStream_t stream) {
    const float* u   = (const float*)d_in[0];
    const float* x   = (const float*)d_in[1];
    const float* Wk  = (const float*)d_in[2];
    // d_in[3] = bk : softmax-invariant, intentionally unused
    const float* Wq  = (const float*)d_in[4];
    const float* bq  = (const float*)d_in[5];
    const float* Wv  = (const float*)d_in[6];
    const float* bv  = (const float*)d_in[7];
    const float* g1  = (const float*)d_in[8];
    const float* be1 = (const float*)d_in[9];
    const float* Wm1 = (const float*)d_in[10];
    const float* bm1 = (const float*)d_in[11];
    const float* Wm2 = (const float*)d_in[12];
    const float* bm2 = (const float*)d_in[13];
    const float* g2  = (const float*)d_in[14];
    const float* be2 = (const float*)d_in[15];
    const float* Ws  = (const float*)d_in[16];
    const float* bs  = (const float*)d_in[17];

    // workspace layout (floats): transposed weights, z, shortcut  (~29.6 MB)
    float* ws    = (float*)d_ws;
    float* WqT   = ws;                 // 64x64
    float* WkT   = ws + 4096;          // 64x64
    float* WvT   = ws + 8192;          // 64x64
    float* WsT   = ws + 12288;         // 64x64
    float* Wm1T  = ws + 16384;         // 256x64
    float* Wm2T  = ws + 32768;         // 64x256
    float* zbuf  = ws + 49152;         // 8*7168*64
    float* shcut = zbuf + 3670016;     // 8*7168*64

    k_transpose<<<16, 256, 0, stream>>>(Wq,  WqT,  64, 64);
    k_transpose<<<16, 256, 0, stream>>>(Wk,  WkT,  64, 64);
    k_transpose<<<16, 256, 0, stream>>>(Wv,  WvT,  64, 64);
    k_transpose<<<16, 256, 0, stream>>>(Ws,  WsT,  64, 64);
    k_transpose<<<64, 256, 0, stream>>>(Wm1, Wm1T, 64, 256);
    k_transpose<<<64, 256, 0, stream>>>(Wm2, Wm2T, 256, 64);

    k_attn<<<3584, 256, 0, stream>>>(u, x, WkT, WqT, bq, WvT, bv, WsT, bs,
                                     zbuf, shcut);
    k_mlp<<<3584, 256, 0, stream>>>(zbuf, shcut, g1, be1, Wm1T, bm1,
                                    Wm2T, bm2, g2, be2, (float*)d_out);
}